// NEW_9259949490823
// MI455X (gfx1250) — compile-verified
//
#include <hip/hip_runtime.h>
#include <hip/hip_bf16.h>
#include <math.h>

// ---------------------------------------------------------------------------
// Swin window attention (W-MSA + shift mask) for gfx1250 (MI455X).
// All GEMMs on V_WMMA_F32_16X16X4_F32 (f32 keeps reference numerics).
// One wave32 per (window, head); multi-accumulator tiling so each K-step
// issues 2-8 independent WMMAs off one shared fragment load.
// X-tile staging uses GLOBAL_LOAD_ASYNC_TO_LDS_B128 when available.
// ---------------------------------------------------------------------------

typedef __attribute__((ext_vector_type(2))) float v2f;
typedef __attribute__((ext_vector_type(8))) float v8f;
typedef __attribute__((ext_vector_type(4))) int   v4i_;

#define WS7_N    49      // tokens per window
#define DIM      128
#define HEADS    4
#define HDIM     32
#define NW_MASK  63      // nW = 64 -> mask index = b & 63
#define NEG_BIG  (-1.0e30f)

// LDS strides padded for conflict-free fragment access (64 banks x 4B)
#define XSTR     132     // X tile row stride   (132 % 64 -> 4-bank step)
#define QSTR     36      // Q/K/V row stride    (36*lm mod 64 all distinct)
#define SSTR     68      // S tile row stride

#define AS_GLOBAL __attribute__((address_space(1)))
#define AS_LDS    __attribute__((address_space(3)))

#if __has_builtin(__builtin_amdgcn_global_load_async_to_lds_b128)
#define HAVE_ASYNC_LDS 1
#else
#define HAVE_ASYNC_LDS 0
#endif

__device__ __forceinline__ void wait_asynccnt0() {
#if __has_builtin(__builtin_amdgcn_s_wait_asynccnt)
  __builtin_amdgcn_s_wait_asynccnt(0);
#else
  asm volatile("s_wait_asynccnt 0x0" ::: "memory");
#endif
}

__device__ __forceinline__ v8f wmma4f32(v2f a, v2f b, v8f c) {
  // (neg_a, A, neg_b, B, c_mod, C, reuse_a, reuse_b)
  return __builtin_amdgcn_wmma_f32_16x16x4_f32(
      false, a, false, b, (short)0, c, false, false);
}

// ---------------------------------------------------------------------------
// Kernel 1: per (window b, head h) fused QKV + attention + O.
//   grid = (4096, 4), block = 32 threads (one wave32; EXEC all-ones at WMMA).
// LDS:
//   qkv_lds : 3 matrices [64][QSTR]           (27648 B)
//   scratch : phase1 = X tile [64][XSTR]      (33792 B)
//             phase2+= S tile [64][SSTR]      (17408 B, overlaps X)
// ---------------------------------------------------------------------------
__global__ __launch_bounds__(32)
void swin_attn_fused_kernel(const float* __restrict__ x,        // (4096,49,128)
                            const float* __restrict__ qkv_w,    // (128,384)
                            const float* __restrict__ qkv_b,    // (384)
                            const float* __restrict__ bias_tab, // (169,4)
                            const float* __restrict__ mask,     // (64,49,49)
                            const int*   __restrict__ rel_idx,  // (49,49)
                            float*       __restrict__ o_ws)     // (200704,128)
{
  __shared__ float qkv_lds[3 * 64 * QSTR];
  __shared__ float scratch[64 * XSTR];

  float* x_lds = scratch;                 // phase 1
  float* s_lds = scratch;                 // phase 2+ (X no longer needed)

  const int lane = threadIdx.x;           // 0..31
  const int half = lane >> 4;             // K-half selector for f32 WMMA frags
  const int lm   = lane & 15;
  const int b    = blockIdx.x;            // window
  const int h    = blockIdx.y;            // head
  const float scale = 0.17677669529663687f;   // 32^-0.5

  const float* maskp = mask + (size_t)(b & NW_MASK) * (WS7_N * WS7_N);

  // ---- Stage X_b (49x128, zero-padded to 64 rows) into LDS ----------------
  // One row per iteration; 32 lanes * float4 = one 128-col row per issue.
#if HAVE_ASYNC_LDS
  for (int row = 0; row < WS7_N; ++row) {
    const float* gsrc = x + ((size_t)b * WS7_N + row) * DIM + lane * 4;
    __builtin_amdgcn_global_load_async_to_lds_b128(
        (AS_GLOBAL v4i_*)gsrc,
        (AS_LDS v4i_*)&x_lds[row * XSTR + lane * 4],
        0, 0);
  }
  for (int row = WS7_N; row < 64; ++row) {   // zero-pad tail rows
    *(float4*)&x_lds[row * XSTR + lane * 4] = make_float4(0.f, 0.f, 0.f, 0.f);
  }
  wait_asynccnt0();
#else
  for (int i = lane; i < 64 * 32; i += 32) {
    const int row = i >> 5;               // 0..63, uniform across the wave
    const int c4  = (i & 31) << 2;        // float4 column
    float4 v = make_float4(0.f, 0.f, 0.f, 0.f);
    if (row < WS7_N)
      v = *(const float4*)(x + ((size_t)b * WS7_N + row) * DIM + c4);
    *(float4*)&x_lds[row * XSTR + c4] = v;
  }
#endif
  __syncthreads();

  // ---- Phase 1: QKV = X_b @ W[:, s*128 + h*32 + *] + bias -----------------
  // 4 M-tile accumulators share each B fragment (global qkv_w, L2-resident).
  for (int s = 0; s < 3; ++s) {
    for (int nt = 0; nt < 2; ++nt) {
      const int ncol = nt * 16 + lm;                // 0..31 (head dim)
      const int wcol = s * DIM + h * HDIM + ncol;   // qkv_w column
      v8f acc[4] = {{}, {}, {}, {}};
#pragma unroll
      for (int k0 = 0; k0 < DIM; k0 += 4) {
        const int ka = k0 + half * 2;
        v2f bb;
        bb.x = qkv_w[(size_t)ka * 384 + wcol];
        bb.y = qkv_w[(size_t)(ka + 1) * 384 + wcol];
#pragma unroll
        for (int mt = 0; mt < 4; ++mt) {
          v2f a;
          a.x = x_lds[(mt * 16 + lm) * XSTR + ka];
          a.y = x_lds[(mt * 16 + lm) * XSTR + ka + 1];
          acc[mt] = wmma4f32(a, bb, acc[mt]);
        }
      }
      const float bias = qkv_b[wcol];
#pragma unroll
      for (int mt = 0; mt < 4; ++mt)
#pragma unroll
        for (int r = 0; r < 8; ++r) {
          const int mm = mt * 16 + r + half * 8;    // C layout: M = r / r+8
          qkv_lds[s * (64 * QSTR) + mm * QSTR + ncol] = acc[mt][r] + bias;
        }
    }
  }
  __syncthreads();

  const float* Q = qkv_lds;
  const float* K = qkv_lds + 64 * QSTR;
  const float* V = qkv_lds + 2 * 64 * QSTR;

  // ---- Phase 2: S = Q @ K^T * scale + rel-bias + mask ---------------------
  // 4 N-tile accumulators share each A (Q) fragment.
  for (int mt = 0; mt < 4; ++mt) {
    const int arow = (mt * 16 + lm) * QSTR;
    v8f acc[4] = {{}, {}, {}, {}};
#pragma unroll
    for (int k0 = 0; k0 < HDIM; k0 += 4) {
      const int ka = k0 + half * 2;
      v2f a;
      a.x = Q[arow + ka];
      a.y = Q[arow + ka + 1];
#pragma unroll
      for (int nt = 0; nt < 4; ++nt) {
        v2f bb;                                     // B[k][n] = K[n][k]
        bb.x = K[(nt * 16 + lm) * QSTR + ka];
        bb.y = K[(nt * 16 + lm) * QSTR + ka + 1];
        acc[nt] = wmma4f32(a, bb, acc[nt]);
      }
    }
#pragma unroll
    for (int nt = 0; nt < 4; ++nt) {
      const int nn = nt * 16 + lm;
#pragma unroll
      for (int r = 0; r < 8; ++r) {
        const int mm = mt * 16 + r + half * 8;
        float sv = NEG_BIG;                          // padded rows/cols
        if (mm < WS7_N && nn < WS7_N) {
          const int ri = rel_idx[mm * WS7_N + nn];
          sv = acc[nt][r] * scale + bias_tab[ri * HEADS + h]
             + maskp[mm * WS7_N + nn];
        }
        s_lds[mm * SSTR + nn] = sv;
      }
    }
  }
  __syncthreads();

  // ---- Phase 3: row softmax (2 rows per lane; all-(-1e30) rows -> uniform)
#pragma unroll
  for (int rr = 0; rr < 2; ++rr) {
    const int row = lane + rr * 32;
    float mx = NEG_BIG;
    for (int j = 0; j < 64; ++j) mx = fmaxf(mx, s_lds[row * SSTR + j]);
    float sum = 0.0f;
    for (int j = 0; j < 64; ++j) {
      const float e = __expf(s_lds[row * SSTR + j] - mx);
      s_lds[row * SSTR + j] = e;
      sum += e;
    }
    const float inv = 1.0f / sum;
    for (int j = 0; j < 64; ++j) s_lds[row * SSTR + j] *= inv;
  }
  __syncthreads();

  // ---- Phase 4: O = P @ V ; store head slice of (200704,128) --------------
  for (int mt = 0; mt < 4; ++mt) {
    const int arow = (mt * 16 + lm) * SSTR;
    v8f acc[2] = {{}, {}};
#pragma unroll
    for (int k0 = 0; k0 < 64; k0 += 4) {
      const int ka = k0 + half * 2;                 // token index (K dim)
      v2f a;
      a.x = s_lds[arow + ka];
      a.y = s_lds[arow + ka + 1];
#pragma unroll
      for (int nt = 0; nt < 2; ++nt) {
        v2f bb;
        bb.x = V[ka * QSTR + nt * 16 + lm];
        bb.y = V[(ka + 1) * QSTR + nt * 16 + lm];
        acc[nt] = wmma4f32(a, bb, acc[nt]);
      }
    }
#pragma unroll
    for (int nt = 0; nt < 2; ++nt)
#pragma unroll
      for (int r = 0; r < 8; ++r) {
        const int mm = mt * 16 + r + half * 8;
        if (mm < WS7_N) {
          o_ws[((size_t)b * WS7_N + mm) * DIM + h * HDIM + nt * 16 + lm]
              = acc[nt][r];
        }
      }
  }
}

// ---------------------------------------------------------------------------
// Kernel 2: out = O(200704,128) @ proj_w(128,128) + proj_b.
//   Each wave owns a full 16-row strip (8 N-tiles, 64 acc VGPRs); one A
//   fragment feeds 8 independent WMMAs per K-step. 200704 % 16 == 0.
// ---------------------------------------------------------------------------
__global__ __launch_bounds__(256)
void swin_proj_kernel(const float* __restrict__ o_ws,    // (200704,128)
                      const float* __restrict__ proj_w,  // (128,128)
                      const float* __restrict__ proj_b,  // (128)
                      float*       __restrict__ out)     // (200704,128)
{
  const int lane = threadIdx.x & 31;
  const int wid  = threadIdx.x >> 5;
  const int half = lane >> 4;
  const int lm   = lane & 15;

  const long m0 = ((long)blockIdx.x * 8 + wid) * 16;   // row strip base

  v8f acc[8] = {{}, {}, {}, {}, {}, {}, {}, {}};
#pragma unroll
  for (int k0 = 0; k0 < DIM; k0 += 4) {
    const int ka = k0 + half * 2;
    v2f a;
    a.x = o_ws[(m0 + lm) * DIM + ka];
    a.y = o_ws[(m0 + lm) * DIM + ka + 1];
#pragma unroll
    for (int nt = 0; nt < 8; ++nt) {
      v2f bb;
      bb.x = proj_w[ka * DIM + nt * 16 + lm];
      bb.y = proj_w[(ka + 1) * DIM + nt * 16 + lm];
      acc[nt] = wmma4f32(a, bb, acc[nt]);
    }
  }
#pragma unroll
  for (int nt = 0; nt < 8; ++nt) {
    const float bias = proj_b[nt * 16 + lm];
#pragma unroll
    for (int r = 0; r < 8; ++r) {
      const long mm = m0 + r + half * 8;
      out[mm * DIM + nt * 16 + lm] = acc[nt][r] + bias;
    }
  }
}

// ---------------------------------------------------------------------------
extern "C" void kernel_launch(void* const* d_in, const int* in_sizes, int n_in,
                              void* d_out, int out_size, void* d_ws, size_t ws_size,
                              hipStream_t stream) {
  const float* x        = (const float*)d_in[0];
  const float* qkv_w    = (const float*)d_in[1];
  const float* qkv_b    = (const float*)d_in[2];
  const float* proj_w   = (const float*)d_in[3];
  const float* proj_b   = (const float*)d_in[4];
  const float* bias_tab = (const float*)d_in[5];
  const float* mask     = (const float*)d_in[6];
  const int*   rel_idx  = (const int*)d_in[7];
  float* out  = (float*)d_out;
  float* o_ws = (float*)d_ws;                 // 200704*128 f32 (~98 MB)

  dim3 grid1(4096, HEADS);
  swin_attn_fused_kernel<<<grid1, 32, 0, stream>>>(
      x, qkv_w, qkv_b, bias_tab, mask, rel_idx, o_ws);

  swin_proj_kernel<<<1568, 256, 0, stream>>>(o_ws, proj_w, proj_b, out);
}